// W_SOCNet_67164698575359
// MI455X (gfx1250) — compile-verified
//
#include <hip/hip_runtime.h>
#include <math.h>

typedef float v2f __attribute__((ext_vector_type(2)));
typedef float v8f __attribute__((ext_vector_type(8)));

#define THREADS 256
#define SPT 16        // steps per thread; assumes L == 4096
#define ROWLEN 4096

__device__ __forceinline__ float softplusf(float z) {
    // numerically stable log1p(exp(z))
    return fmaxf(z, 0.0f) + log1pf(expf(-fabsf(z)));
}

__global__ __launch_bounds__(THREADS) void socnet_scan_kernel(
    const float4* __restrict__ X, const float* __restrict__ SC,
    const float* __restrict__ wi1, const float* __restrict__ bi1,
    const float* __restrict__ wi2, const float* __restrict__ bi2,
    const float* __restrict__ we1, const float* __restrict__ be1,
    const float* __restrict__ we2, const float* __restrict__ be2,
    float4* __restrict__ Y)
{
    __shared__ float tot[4][THREADS];   // thread totals, later overwritten with exclusive prefixes
    __shared__ __align__(16) float colTot[4][16];

    const int n = blockIdx.x;
    const int t = threadIdx.x;
    const float4* xrow = X + (size_t)n * ROWLEN;
    float4*       yrow = Y + (size_t)n * ROWLEN;

    // tiny "MLP" params (hidden width 1)
    const float e10 = we1[0], e11 = we1[1], be1v = be1[0];
    const float we2v = we2[0], be2v = be2[0];

    const float Q = SC[n*4+0], eta0 = SC[n*4+1], R = SC[n*4+2], cap = SC[n*4+3];
    const float invQ = 1.0f / (3600.0f * Q);

    // soc_init (redundant per thread; L2-cached, negligible)
    float4 x0 = xrow[0];
    float zi  = wi1[0]*x0.y + wi1[1]*x0.z + wi1[2]*x0.w + wi1[3]*R + bi1[0];
    float soc = cap * (1.0f + (softplusf(zi)*wi2[0] + bi2[0]));

    // ---- Phase 1: per-thread segment: inc + local inclusive sums ----
    float4 li[SPT];
    float sx = 0.f, sy = 0.f, sz = 0.f, sw = 0.f;
    const int l0 = t * SPT;
    float4 xc  = xrow[l0];
    float de_c = softplusf(e10*xc.y + e11*xc.z + be1v)*we2v + be2v;
#pragma unroll
    for (int k = 0; k < SPT; ++k) {
        const int l = l0 + k;
        if (l < ROWLEN - 1) {
            float4 xn  = xrow[l + 1];
            float de_n = softplusf(e10*xn.y + e11*xn.z + be1v)*we2v + be2v;
            float dy   = eta0 * (1.0f + de_c) * xc.y * invQ;
            sx += (xn.y - xc.y) * dy;
            sy += (xn.z - xc.z) * dy;
            sz += (xn.w - xc.w) * dy;
            sw += (de_n - de_c) * dy;
            xc = xn; de_c = de_n;
        }
        li[k] = make_float4(sx, sy, sz, sw);
    }
    tot[0][t] = sx; tot[1][t] = sy; tot[2][t] = sz; tot[3][t] = sw;
    __syncthreads();

    // ---- Phase 2: block scan of 256 totals per channel via WMMA matmul-scan ----
    // M[i][j] = tot[ch][j*16 + i]; inclusive-in-column scan S = L16 x M with
    // L16 = lower-triangular ones, as 4 chained v_wmma_f32_16x16x4_f32 (K split 4x4).
    const int wave = t >> 5;
    const int lane = t & 31;
    if (wave < 4) {                       // wave-uniform branch: EXEC stays all-1s for WMMA
        const int ch = wave;
        const int j  = lane & 15;               // column (B/D N index)
        const int i  = lane & 15;               // A-matrix M row this lane feeds
        const int kRow0 = (lane < 16) ? 0 : 2;  // ISA 16x4 f32 A layout: lanes16-31 hold K=2,3
        const int r0    = (lane < 16) ? 0 : 8;  // D rows: lanes16-31 hold M=8..15
        v8f acc = {0.f,0.f,0.f,0.f,0.f,0.f,0.f,0.f};
#pragma unroll
        for (int kk = 0; kk < 4; ++kk) {
            const int k0 = 4*kk + kRow0;        // global K row for VGPR0; VGPR1 is k0+1
            v2f a, b;
            a.x = (k0     <= i) ? 1.0f : 0.0f;  // L16[i][k0]
            a.y = (k0 + 1 <= i) ? 1.0f : 0.0f;  // L16[i][k0+1]
            b.x = tot[ch][j*16 + k0];           // M[k0][j]
            b.y = tot[ch][j*16 + k0 + 1];       // M[k0+1][j]
            acc = __builtin_amdgcn_wmma_f32_16x16x4_f32(
                /*neg_a=*/false, a, /*neg_b=*/false, b,
                /*c_mod=*/(short)0, acc, /*reuse_a=*/false, /*reuse_b=*/false);
        }
        // column totals = S[15][j], held by lanes 16..31 in acc[7]
        if (lane >= 16) colTot[ch][j] = acc[7];
        __builtin_amdgcn_wave_barrier();        // same-wave DS ordering; fence the compiler

        // Unconditional b128 loads of all 16 column totals, then masked accumulate:
        // keeps this branch-free (v_cndmask on registers, no divergent ds_loads).
        const float4* ctv = (const float4*)&colTot[ch][0];
        float4 c0 = ctv[0], c1 = ctv[1], c2 = ctv[2], c3 = ctv[3];
        float cv[16] = { c0.x,c0.y,c0.z,c0.w, c1.x,c1.y,c1.z,c1.w,
                         c2.x,c2.y,c2.z,c2.w, c3.x,c3.y,c3.z,c3.w };
        float coloff = 0.0f;
#pragma unroll
        for (int jj = 0; jj < 16; ++jj)
            coloff += (jj < j) ? cv[jj] : 0.0f;

        // exclusive prefix per thread index tt = j*16 + row, written in place
#pragma unroll
        for (int r = 0; r < 8; ++r) {
            const int tt = j*16 + (r0 + r);
            tot[ch][tt] = acc[r] + coloff - tot[ch][tt];
        }
    }
    __syncthreads();

    // ---- Phase 3: stream outputs ----
    const float4 base = make_float4(soc + tot[0][t], soc + tot[1][t],
                                    soc + tot[2][t], soc + tot[3][t]);
    if (t == 0) yrow[0] = make_float4(soc, soc, soc, soc);
#pragma unroll
    for (int k = 0; k < SPT; ++k) {
        const int l = l0 + k;
        if (l < ROWLEN - 1) {
            yrow[l + 1] = make_float4(base.x + li[k].x, base.y + li[k].y,
                                      base.z + li[k].z, base.w + li[k].w);
        }
    }
}

extern "C" void kernel_launch(void* const* d_in, const int* in_sizes, int n_in,
                              void* d_out, int out_size, void* d_ws, size_t ws_size,
                              hipStream_t stream) {
    (void)n_in; (void)out_size; (void)d_ws; (void)ws_size;
    const float4* X  = (const float4*)d_in[0];
    const float*  SC = (const float*)d_in[1];
    const float* wi1 = (const float*)d_in[2];
    const float* bi1 = (const float*)d_in[3];
    const float* wi2 = (const float*)d_in[4];
    const float* bi2 = (const float*)d_in[5];
    const float* we1 = (const float*)d_in[6];
    const float* be1 = (const float*)d_in[7];
    const float* we2 = (const float*)d_in[8];
    const float* be2 = (const float*)d_in[9];
    float4* Y = (float4*)d_out;

    const int N = in_sizes[1] / 4;   // SC is (N,4); L fixed at 4096 per reference
    socnet_scan_kernel<<<dim3(N), dim3(THREADS), 0, stream>>>(
        X, SC, wi1, bi1, wi2, bi2, we1, be1, we2, be2, Y);
}